// BoundLoss_59820304499339
// MI455X (gfx1250) — compile-verified
//
#include <hip/hip_runtime.h>
#include <math.h>

// Problem constants (match reference)
#define BB   8
#define CCH  8
#define HH   640
#define WW   640
#define NN   (HH * WW)      // 409600 pixels per image
#define MI   16             // instances per image
#define EPSF 1e-12f

typedef __attribute__((ext_vector_type(2))) float v2f;
typedef __attribute__((ext_vector_type(8))) float v8f;

// -------- workspace layout (floats) --------
// [0,512)     sumK  : per (b,m) 4-dim kernel-pixel embedding sums
// [512,640)   cntK  : per (b,m) kernel-pixel counts
// [640,1152)  G     : finalized centroids
// [1152,1280) valid : 1.0 if instance valid (cnt>0 && m>=1)
// [1280,1408) sumT  : per (b,m) pull-loss sums over text pixels
// [1408,1536) cntT  : per (b,m) text-pixel counts
#define WS_FLOATS 1536

__global__ void k_zero(float* __restrict__ ws, int n) {
    int i = blockIdx.x * blockDim.x + threadIdx.x;
    if (i < n) ws[i] = 0.0f;
}

// ---------------- Pass 1: centroid accumulation (streaming, HBM-bound) ----------------
__global__ void k_centroid_accum(const float* __restrict__ outp,
                                 const int* __restrict__ gk,
                                 float* __restrict__ sumK,
                                 float* __restrict__ cntK) {
    __shared__ float sS[8][MI][4];   // per-wave embedding sums
    __shared__ float sC[8][MI];      // per-wave counts
    const int tid = threadIdx.x;
    const int wv  = tid >> 5;
    const int b   = blockIdx.y;

    for (int i = tid; i < 8 * MI * 4; i += blockDim.x) ((float*)sS)[i] = 0.0f;
    for (int i = tid; i < 8 * MI;     i += blockDim.x) ((float*)sC)[i] = 0.0f;
    __syncthreads();

    const float4* c0 = (const float4*)(outp + ((size_t)b * CCH + 4) * NN);
    const float4* c1 = (const float4*)(outp + ((size_t)b * CCH + 5) * NN);
    const float4* c2 = (const float4*)(outp + ((size_t)b * CCH + 6) * NN);
    const float4* c3 = (const float4*)(outp + ((size_t)b * CCH + 7) * NN);
    const int4*   kk = (const int4*)(gk + (size_t)b * NN);

    const int ngrp   = NN / 4;
    const int stride = gridDim.x * blockDim.x;
    for (int g = blockIdx.x * blockDim.x + tid; g < ngrp; g += stride) {
        __builtin_prefetch(&c0[g + stride], 0, 0);   // global_prefetch_b8
        __builtin_prefetch(&kk[g + stride], 0, 0);
        float4 s0 = c0[g], s1 = c1[g], s2 = c2[g], s3 = c3[g];
        int4 id = kk[g];
#define ACC1(IDX, CMP)                                            \
        {   int m = IDX;                                          \
            atomicAdd(&sS[wv][m][0], s0.CMP);                     \
            atomicAdd(&sS[wv][m][1], s1.CMP);                     \
            atomicAdd(&sS[wv][m][2], s2.CMP);                     \
            atomicAdd(&sS[wv][m][3], s3.CMP);                     \
            atomicAdd(&sC[wv][m], 1.0f); }
        ACC1(id.x, x) ACC1(id.y, y) ACC1(id.z, z) ACC1(id.w, w)
#undef ACC1
    }
    __syncthreads();

    // cross-wave reduce, then one global atomic per (m,component)
    if (tid < MI * 5) {
        int m = tid / 5, c = tid % 5;
        float s = 0.0f;
        if (c < 4) {
            for (int q = 0; q < 8; ++q) s += sS[q][m][c];
            atomicAdd(&sumK[((size_t)b * MI + m) * 4 + c], s);
        } else {
            for (int q = 0; q < 8; ++q) s += sC[q][m];
            atomicAdd(&cntK[(size_t)b * MI + m], s);
        }
    }
}

// ---------------- Pass 2: finalize centroids + push loss via V_WMMA_F32_16X16X4_F32 ----
// One wave per image. Gram = G (16x4) x G^T (4x16) in a single WMMA per wave;
// pairwise dist^2 = |Gi|^2 + |Gj|^2 - 2*Gram[i][j].
__global__ void k_push_wmma(const float* __restrict__ sumK,
                            const float* __restrict__ cntK,
                            float* __restrict__ Gout,
                            float* __restrict__ validf,
                            float* __restrict__ outPush) {
    __shared__ float sG[BB][MI][4];
    __shared__ float sV[BB][MI];
    __shared__ float sR[BB][32];
    const int tid = threadIdx.x;

    if (tid < BB * MI) {
        int b = tid >> 4, m = tid & 15;
        float cnt = cntK[tid];
        float inv = 1.0f / fmaxf(cnt, 1.0f);
#pragma unroll
        for (int c = 0; c < 4; ++c) {
            float g = sumK[tid * 4 + c] * inv;
            sG[b][m][c] = g;
            Gout[tid * 4 + c] = g;
        }
        float vl = (cnt > 0.0f && m >= 1) ? 1.0f : 0.0f;
        sV[b][m] = vl;
        validf[tid] = vl;
    }
    __syncthreads();

    const int wv = tid >> 5;        // image index
    const int lane = tid & 31;
    const int n = lane & 15;        // column j of Gram held by this lane
    const int half = lane >> 4;

    // A-matrix 16x4 f32 layout: lane%16 = row M; VGPR0 = K 0/2, VGPR1 = K 1/3 by half-wave.
    // B = G^T has the same per-lane register image (Gram is symmetric).
    v2f a;
    a.x = sG[wv][n][2 * half];
    a.y = sG[wv][n][2 * half + 1];
    v8f acc = {};
#if __has_builtin(__builtin_amdgcn_wmma_f32_16x16x4_f32)
    acc = __builtin_amdgcn_wmma_f32_16x16x4_f32(
        /*neg_a=*/false, a, /*neg_b=*/false, a,
        /*c_mod=*/(short)0, acc, /*reuse_a=*/false, /*reuse_b=*/false);
#else
    // scalar fallback (should not be taken on gfx1250)
#pragma unroll
    for (int v = 0; v < 8; ++v) {
        int i = v + 8 * half;
        float s = 0.0f;
        for (int c = 0; c < 4; ++c) s += sG[wv][i][c] * sG[wv][n][c];
        acc[v] = s;
    }
#endif

    float nj = 0.0f;
#pragma unroll
    for (int c = 0; c < 4; ++c) nj += sG[wv][n][c] * sG[wv][n][c];
    const float vj = sV[wv][n];

    float part = 0.0f;
#pragma unroll
    for (int v = 0; v < 8; ++v) {
        int i = v + 8 * half;           // row of Gram in acc[v] (D-matrix layout)
        float ni = 0.0f;
        for (int c = 0; c < 4; ++c) ni += sG[wv][i][c] * sG[wv][i][c];
        float d2 = fmaxf(ni + nj - 2.0f * acc[v], 0.0f);
        float dk = sqrtf(d2 + EPSF);
        float t  = fmaxf(3.0f - dk, 0.0f);
        float val = log1pf(t * t);
        if (i < n) part += val * sV[wv][i] * vj;   // upper triangle i<j, both valid
    }
    sR[wv][lane] = part;
    __syncthreads();

    if (tid < BB) {
        float s = 0.0f;
        for (int l = 0; l < 32; ++l) s += sR[tid][l];
        float nv = 0.0f;
        for (int m = 0; m < MI; ++m) nv += sV[tid][m];
        float denom = fmaxf(nv * (nv - 1.0f), 1.0f);
        outPush[tid] = (nv > 1.0f) ? (s / denom) : 0.0f;
    }
}

// ---------------- Pass 3: pull-loss accumulation (streaming, HBM-bound) ---------------
__global__ void k_pull_accum(const float* __restrict__ outp,
                             const int* __restrict__ gt,
                             const float* __restrict__ G,
                             float* __restrict__ sumT,
                             float* __restrict__ cntT) {
    __shared__ float sG[MI][4];
    __shared__ float sS[8][MI];
    __shared__ float sC[8][MI];
    const int tid = threadIdx.x;
    const int wv  = tid >> 5;
    const int b   = blockIdx.y;

    if (tid < MI * 4) ((float*)sG)[tid] = G[(size_t)b * MI * 4 + tid];
    for (int i = tid; i < 8 * MI; i += blockDim.x) {
        ((float*)sS)[i] = 0.0f;
        ((float*)sC)[i] = 0.0f;
    }
    __syncthreads();

    const float4* c0 = (const float4*)(outp + ((size_t)b * CCH + 4) * NN);
    const float4* c1 = (const float4*)(outp + ((size_t)b * CCH + 5) * NN);
    const float4* c2 = (const float4*)(outp + ((size_t)b * CCH + 6) * NN);
    const float4* c3 = (const float4*)(outp + ((size_t)b * CCH + 7) * NN);
    const int4*   tt = (const int4*)(gt + (size_t)b * NN);

    const int ngrp   = NN / 4;
    const int stride = gridDim.x * blockDim.x;
    for (int g = blockIdx.x * blockDim.x + tid; g < ngrp; g += stride) {
        __builtin_prefetch(&c0[g + stride], 0, 0);
        __builtin_prefetch(&tt[g + stride], 0, 0);
        float4 s0 = c0[g], s1 = c1[g], s2 = c2[g], s3 = c3[g];
        int4 id = tt[g];
#define PULL1(IDX, CMP)                                               \
        {   int m = IDX;                                              \
            float d0 = s0.CMP - sG[m][0];                             \
            float d1 = s1.CMP - sG[m][1];                             \
            float d2 = s2.CMP - sG[m][2];                             \
            float d3 = s3.CMP - sG[m][3];                             \
            float dd = d0*d0 + d1*d1 + d2*d2 + d3*d3;                 \
            float d  = sqrtf(dd + EPSF) - 0.5f;                       \
            float t  = fmaxf(d, 0.0f);                                \
            float l  = log1pf(t * t);                                 \
            atomicAdd(&sS[wv][m], l);                                 \
            atomicAdd(&sC[wv][m], 1.0f); }
        PULL1(id.x, x) PULL1(id.y, y) PULL1(id.z, z) PULL1(id.w, w)
#undef PULL1
    }
    __syncthreads();

    if (tid < MI * 2) {
        int m = tid >> 1;
        float s = 0.0f;
        if ((tid & 1) == 0) {
            for (int q = 0; q < 8; ++q) s += sS[q][m];
            atomicAdd(&sumT[(size_t)b * MI + m], s);
        } else {
            for (int q = 0; q < 8; ++q) s += sC[q][m];
            atomicAdd(&cntT[(size_t)b * MI + m], s);
        }
    }
}

// ---------------- Pass 4: finalize pull loss ----------------
__global__ void k_finalize_pull(const float* __restrict__ sumT,
                                const float* __restrict__ cntT,
                                const float* __restrict__ validf,
                                float* __restrict__ outPull) {
    int b = threadIdx.x;
    if (b < BB) {
        float nv = 0.0f, acc = 0.0f;
        for (int m = 0; m < MI; ++m) {
            float v = validf[b * MI + m];
            nv += v;
            acc += v * sumT[b * MI + m] / fmaxf(cntT[b * MI + m], 1.0f);
        }
        outPull[b] = acc / fmaxf(nv, 1.0f);
    }
}

extern "C" void kernel_launch(void* const* d_in, const int* in_sizes, int n_in,
                              void* d_out, int out_size, void* d_ws, size_t ws_size,
                              hipStream_t stream) {
    const float* outp = (const float*)d_in[0];   // [8,8,640,640] f32
    const int*   gt   = (const int*)d_in[1];     // gt_texts   [8,640,640] i32
    const int*   gk   = (const int*)d_in[2];     // gt_kernels [8,640,640] i32
    // d_in[3], d_in[4] (gt_tops/gt_bots) do not contribute to the result.

    float* ws    = (float*)d_ws;
    float* sumK  = ws;
    float* cntK  = ws + 512;
    float* G     = ws + 640;
    float* vldf  = ws + 1152;
    float* sumT  = ws + 1280;
    float* cntT  = ws + 1408;
    float* out   = (float*)d_out;   // [0..7] pull, [8..15] push

    hipLaunchKernelGGL(k_zero, dim3((WS_FLOATS + 255) / 256), dim3(256), 0, stream,
                       ws, WS_FLOATS);

    dim3 gridS(64, BB);   // grid-stride streaming, 512 blocks total
    hipLaunchKernelGGL(k_centroid_accum, gridS, dim3(256), 0, stream,
                       outp, gk, sumK, cntK);
    hipLaunchKernelGGL(k_push_wmma, dim3(1), dim3(256), 0, stream,
                       sumK, cntK, G, vldf, out + 8);
    hipLaunchKernelGGL(k_pull_accum, gridS, dim3(256), 0, stream,
                       outp, gt, G, sumT, cntT);
    hipLaunchKernelGGL(k_finalize_pull, dim3(1), dim3(64), 0, stream,
                       sumT, cntT, vldf, out);
}